// GATModel_10479720202337
// MI455X (gfx1250) — compile-verified
//
#include <hip/hip_runtime.h>
#include <cstdint>
#include <cstddef>

#define FIN       128
#define HID       256
#define HEADS     4
#define CH        64
#define BATCH     4096
#define NEG_SLOPE 0.2f
#define BN_EPS    1e-5f

typedef float v2f __attribute__((ext_vector_type(2)));
typedef float v8f __attribute__((ext_vector_type(8)));

// ---------------- fills ----------------
__global__ void fill_f32(float* __restrict__ p, float v, int n) {
    int i = blockIdx.x * blockDim.x + threadIdx.x;
    if (i < n) p[i] = v;
}
__global__ void fill_u32(unsigned* __restrict__ p, unsigned v, int n) {
    int i = blockIdx.x * blockDim.x + threadIdx.x;
    if (i < n) p[i] = v;
}

// ---------------- order-preserving float<->uint for atomicMax ----------------
__device__ __forceinline__ unsigned f2ord(float f) {
    unsigned b = __float_as_uint(f);
    return (b & 0x80000000u) ? ~b : (b | 0x80000000u);
}
__device__ __forceinline__ float ord2f(unsigned u) {
    return (u & 0x80000000u) ? __uint_as_float(u & 0x7FFFFFFFu) : __uint_as_float(~u);
}

// ---------------- f32 GEMM: C[M x NC] = A[M x KK] @ B[KK x NC], row-major ----------------
// NC, KK compile-time -> all load addresses are fixed bases + immediate offsets.
// One wave -> 32x16 output (two 16x16 WMMA tiles sharing one B fragment).
// 8 waves/block -> 32x128 block tile. grid = (NC/128, ceil(M/32)), block = 256.
template <int NC, int KK>
__global__ void gemm_wmma_f32(const float* __restrict__ A, const float* __restrict__ B,
                              float* __restrict__ C, int M)
{
    const int lane = threadIdx.x & 31;
    const int wave = threadIdx.x >> 5;
    const int m0 = blockIdx.y << 5;                  // 32 rows per block
    const int n0 = (blockIdx.x << 7) + (wave << 4);  // 8 waves * 16 cols
    if (m0 >= M) return;

    const int half = lane >> 4;      // 0: lanes 0-15 (K=0,1) | 1: lanes 16-31 (K=2,3)
    const int l15  = lane & 15;
    int ar0 = m0 + l15;      if (ar0 >= M) ar0 = M - 1;   // clamp: EXEC stays all-ones
    int ar1 = m0 + 16 + l15; if (ar1 >= M) ar1 = M - 1;
    const int bcol = n0 + l15;

    const float* ap0 = A + (size_t)ar0 * KK + (half << 1);
    const float* ap1 = A + (size_t)ar1 * KK + (half << 1);
    const float* bp  = B + (half << 1) * NC + bcol;

    v8f acc0 = {}, acc1 = {};
    #pragma unroll 8
    for (int k = 0; k < KK; k += 4) {
        v2f a0 = *(const v2f*)(ap0 + k);     // 8B-aligned contiguous K pair, imm offset
        v2f a1 = *(const v2f*)(ap1 + k);
        v2f b;
        b.x = bp[k * NC];                    // imm offsets (NC, KK compile-time)
        b.y = bp[k * NC + NC];
        acc0 = __builtin_amdgcn_wmma_f32_16x16x4_f32(
                   false, a0, false, b, (short)0, acc0, false, false);
        acc1 = __builtin_amdgcn_wmma_f32_16x16x4_f32(
                   false, a1, false, b, (short)0, acc1, false, false);
    }
    #pragma unroll
    for (int v = 0; v < 8; ++v) {                 // VGPR v: lanes0-15 M=v, lanes16-31 M=v+8
        const int row0 = m0 + v + (half << 3);
        const int row1 = row0 + 16;
        if (row0 < M) C[(size_t)row0 * NC + n0 + l15] = acc0[v];
        if (row1 < M) C[(size_t)row1 * NC + n0 + l15] = acc1[v];
    }
}

// ---------------- attention scores: s_src/s_dst[n,h] = sum_c xl[n,h,c]*a[h,c] ----------------
// wave per node; lane l covers cols [8l, 8l+8) (one head per 8-lane group).
__global__ void att_scores(const float* __restrict__ XL,
                           const float* __restrict__ a_src, const float* __restrict__ a_dst,
                           float* __restrict__ ssrc, float* __restrict__ sdst, int N)
{
    const int lane = threadIdx.x & 31;
    const int node = (blockIdx.x * blockDim.x + threadIdx.x) >> 5;
    if (node >= N) return;
    const int head = lane >> 3;
    const int boff = head * CH + (lane & 7) * 8;
    const float* row = XL + (size_t)node * HID + lane * 8;
    float s1 = 0.f, s2 = 0.f;
    #pragma unroll
    for (int j = 0; j < 8; ++j) {
        float x = row[j];
        s1 += x * a_src[boff + j];
        s2 += x * a_dst[boff + j];
    }
    #pragma unroll
    for (int m = 1; m < 8; m <<= 1) {            // reduce within 8-lane head group
        s1 += __shfl_xor(s1, m, 32);
        s2 += __shfl_xor(s2, m, 32);
    }
    if ((lane & 7) == 0) {
        ssrc[node * HEADS + head] = s1;
        sdst[node * HEADS + head] = s2;
    }
}

// ---------------- edge pass 1: leaky-relu logits + segment max (atomicMax) ----------------
__global__ void edge_logit_max(const long long* __restrict__ e_src, const long long* __restrict__ e_dst,
                               const float* __restrict__ ssrc, const float* __restrict__ sdst,
                               float* __restrict__ elog, unsigned* __restrict__ emax, int E, int N)
{
    const int t = blockIdx.x * blockDim.x + threadIdx.x;
    const int ET = E + N;
    if (t >= ET) return;
    int s, d;
    if (t < E) { s = (int)e_src[t]; d = (int)e_dst[t]; } else { s = d = t - E; }
    #pragma unroll
    for (int h = 0; h < HEADS; ++h) {
        float v = ssrc[s * HEADS + h] + sdst[d * HEADS + h];
        v = (v > 0.f) ? v : NEG_SLOPE * v;
        elog[(size_t)t * HEADS + h] = v;
        atomicMax(&emax[d * HEADS + h], f2ord(v));
    }
}

// ---------------- edge pass 2: w = exp(e - max[dst]); denom[dst] += w ----------------
__global__ void edge_exp_sum(const long long* __restrict__ e_src, const long long* __restrict__ e_dst,
                             float* __restrict__ elog, const unsigned* __restrict__ emax,
                             float* __restrict__ denom, int E, int N)
{
    const int t = blockIdx.x * blockDim.x + threadIdx.x;
    const int ET = E + N;
    if (t >= ET) return;
    int d;
    if (t < E) { d = (int)e_dst[t]; } else { d = t - E; }
    #pragma unroll
    for (int h = 0; h < HEADS; ++h) {
        float m = ord2f(emax[d * HEADS + h]);
        float w = expf(elog[(size_t)t * HEADS + h] - m);
        elog[(size_t)t * HEADS + h] = w;
        atomicAdd(&denom[d * HEADS + h], w);
    }
}

// ---------------- edge pass 3: out[dst] += (w/denom[dst]) * xl[src] ----------------
// wave per edge; lane l handles cols [8l, 8l+8) (head = l/8). Coalesced 1KB row reads,
// native GLOBAL_ATOMIC_ADD_F32 scatter.
__global__ void edge_aggregate(const long long* __restrict__ e_src, const long long* __restrict__ e_dst,
                               const float* __restrict__ XL, const float* __restrict__ elog,
                               const float* __restrict__ denom, float* __restrict__ AG, int E, int N)
{
    const int lane = threadIdx.x & 31;
    const int t = (blockIdx.x * blockDim.x + threadIdx.x) >> 5;
    const int ET = E + N;
    if (t >= ET) return;
    int s, d;
    if (t < E) { s = (int)e_src[t]; d = (int)e_dst[t]; } else { s = d = t - E; }
    const int head = lane >> 3;
    const float coeff = elog[(size_t)t * HEADS + head] / denom[d * HEADS + head];
    const float* xr = XL + (size_t)s * HID + lane * 8;
    float* orow     = AG + (size_t)d * HID + lane * 8;
    #pragma unroll
    for (int j = 0; j < 8; ++j)
        atomicAdd(&orow[j], coeff * xr[j]);
}

// ---------------- BN column statistics (thread=column, block=row slab) ----------------
__global__ void col_stats(const float* __restrict__ Z, const float* __restrict__ bias,
                          float* __restrict__ csum, float* __restrict__ csumsq,
                          int M, int C, int pre_relu, int rows_per_block)
{
    const int c = threadIdx.x;
    if (c >= C) return;
    const int r0 = blockIdx.x * rows_per_block;
    int r1 = r0 + rows_per_block; if (r1 > M) r1 = M;
    const float b = bias[c];
    float s = 0.f, s2 = 0.f;
    for (int r = r0; r < r1; ++r) {
        float v = Z[(size_t)r * C + c] + b;
        if (pre_relu) v = fmaxf(v, 0.f);
        s += v; s2 += v * v;
    }
    atomicAdd(&csum[c], s);
    atomicAdd(&csumsq[c], s2);
}

__global__ void bn_finalize(const float* __restrict__ csum, const float* __restrict__ csumsq,
                            float* __restrict__ mean, float* __restrict__ rstd, int M, int C)
{
    const int c = blockIdx.x * blockDim.x + threadIdx.x;
    if (c >= C) return;
    float m = csum[c] / (float)M;
    float var = csumsq[c] / (float)M - m * m;
    mean[c] = m;
    rstd[c] = rsqrtf(var + BN_EPS);
}

// ---------------- fused bias + (relu?) + BN + (relu?) apply ----------------
__global__ void bn_apply(const float* __restrict__ Z, const float* __restrict__ bias,
                         const float* __restrict__ gamma, const float* __restrict__ beta,
                         const float* __restrict__ mean, const float* __restrict__ rstd,
                         float* __restrict__ Y, int M, int C, int pre_relu, int post_relu)
{
    const int i = blockIdx.x * blockDim.x + threadIdx.x;
    if (i >= M * C) return;
    const int c = i % C;
    float v = Z[i] + bias[c];
    if (pre_relu) v = fmaxf(v, 0.f);
    float y = gamma[c] * (v - mean[c]) * rstd[c] + beta[c];
    if (post_relu) y = fmaxf(y, 0.f);
    Y[i] = y;
}

// ---------------- final 128 -> 1 projection: wave per row ----------------
__global__ void final_dot(const float* __restrict__ Z, const float* __restrict__ W,
                          const float* __restrict__ b, float* __restrict__ out, int M)
{
    const int lane = threadIdx.x & 31;
    const int row = (blockIdx.x * blockDim.x + threadIdx.x) >> 5;
    if (row >= M) return;
    float s = 0.f;
    #pragma unroll
    for (int j = 0; j < 4; ++j) {
        int c = lane * 4 + j;
        s += Z[(size_t)row * 128 + c] * W[c];
    }
    #pragma unroll
    for (int m = 16; m > 0; m >>= 1) s += __shfl_xor(s, m, 32);
    if (lane == 0) out[row] = s + b[0];
}

// =======================================================================================
extern "C" void kernel_launch(void* const* d_in, const int* in_sizes, int n_in,
                              void* d_out, int out_size, void* d_ws, size_t ws_size,
                              hipStream_t stream)
{
    if (n_in < 25) return;
    const float*     x   = (const float*)d_in[0];
    const long long* ei  = (const long long*)d_in[1];   // int64 per reference
    const float* W0  = (const float*)d_in[3];
    const float* as0 = (const float*)d_in[4];
    const float* ad0 = (const float*)d_in[5];
    const float* b0  = (const float*)d_in[6];
    const float* g0  = (const float*)d_in[7];
    const float* be0 = (const float*)d_in[8];
    const float* W1  = (const float*)d_in[9];
    const float* as1 = (const float*)d_in[10];
    const float* ad1 = (const float*)d_in[11];
    const float* b1  = (const float*)d_in[12];
    const float* g1  = (const float*)d_in[13];
    const float* be1 = (const float*)d_in[14];
    const float* Wc1 = (const float*)d_in[15];
    const float* bc1 = (const float*)d_in[16];
    const float* gc1 = (const float*)d_in[17];
    const float* bec1= (const float*)d_in[18];
    const float* Wc2 = (const float*)d_in[19];
    const float* bc2 = (const float*)d_in[20];
    const float* gc2 = (const float*)d_in[21];
    const float* bec2= (const float*)d_in[22];
    const float* Wc3 = (const float*)d_in[23];
    const float* bc3 = (const float*)d_in[24];

    const int N  = in_sizes[0] / FIN;       // 50000
    const int E  = in_sizes[1] / 2;         // 500000
    const int ET = E + N;

    // ---- workspace carve (bump allocator, 256B aligned) ----
    char* w = (char*)d_ws;
    size_t off = 0;
    auto take = [&](size_t bytes) -> void* {
        void* p = w + off;
        off = (off + bytes + 255) & ~(size_t)255;
        return p;
    };
    float*    P      = (float*)   take((size_t)N * HID * 4);   // ping
    float*    Q      = (float*)   take((size_t)N * HID * 4);   // pong
    float*    ssrc   = (float*)   take((size_t)N * HEADS * 4);
    float*    sdst   = (float*)   take((size_t)N * HEADS * 4);
    unsigned* emax   = (unsigned*)take((size_t)N * HEADS * 4);
    float*    denom  = (float*)   take((size_t)N * HEADS * 4);
    float*    elog   = (float*)   take((size_t)ET * HEADS * 4);
    float*    csum   = (float*)   take(HID * 4);
    float*    csumsq = (float*)   take(HID * 4);
    float*    cmean  = (float*)   take(HID * 4);
    float*    crstd  = (float*)   take(HID * 4);
    float*    R      = (float*)   take((size_t)BATCH * 128 * 4);
    if (off > ws_size) return;

    const long long* e_src = ei;
    const long long* e_dst = ei + E;

    auto gat_common = [&](const float* XL, float* AG, float* HOUT,
                          const float* as, const float* ad,
                          const float* bias, const float* gamma, const float* beta)
    {
        // per-node head scores
        att_scores<<<(N * 32 + 255) / 256, 256, 0, stream>>>(XL, as, ad, ssrc, sdst, N);
        // init accumulators
        fill_u32<<<(N * HEADS + 255) / 256, 256, 0, stream>>>(emax, 0u, N * HEADS);
        fill_f32<<<(N * HEADS + 255) / 256, 256, 0, stream>>>(denom, 0.f, N * HEADS);
        fill_f32<<<(N * HID + 255) / 256, 256, 0, stream>>>(AG, 0.f, N * HID);
        // softmax over dst segments (3 passes)
        edge_logit_max<<<(ET + 255) / 256, 256, 0, stream>>>(e_src, e_dst, ssrc, sdst, elog, emax, E, N);
        edge_exp_sum  <<<(ET + 255) / 256, 256, 0, stream>>>(e_src, e_dst, elog, emax, denom, E, N);
        edge_aggregate<<<((size_t)ET * 32 + 255) / 256, 256, 0, stream>>>(e_src, e_dst, XL, elog, denom, AG, E, N);
        // bn(relu(AG + bias)) -> HOUT
        fill_f32<<<1, HID, 0, stream>>>(csum,   0.f, HID);
        fill_f32<<<1, HID, 0, stream>>>(csumsq, 0.f, HID);
        col_stats<<<(N + 63) / 64, HID, 0, stream>>>(AG, bias, csum, csumsq, N, HID, /*pre_relu=*/1, 64);
        bn_finalize<<<1, HID, 0, stream>>>(csum, csumsq, cmean, crstd, N, HID);
        bn_apply<<<(N * HID + 255) / 256, 256, 0, stream>>>(AG, bias, gamma, beta, cmean, crstd,
                                                            HOUT, N, HID, /*pre_relu=*/1, /*post_relu=*/0);
    };

    // Layer 0: x -> h1 (in P)    [XL=P is free after aggregate; apply reads Q, writes P]
    gemm_wmma_f32<HID, FIN><<<dim3(HID / 128, (N + 31) / 32), 256, 0, stream>>>(x, W0, P, N);
    gat_common(/*XL=*/P, /*AG=*/Q, /*HOUT=*/P, as0, ad0, b0, g0, be0);
    // Layer 1: h1(P) -> h2 (in Q) [gemm reads P first; AG=P is only zeroed after gemm completes]
    gemm_wmma_f32<HID, HID><<<dim3(HID / 128, (N + 31) / 32), 256, 0, stream>>>(P, W1, Q, N);
    gat_common(/*XL=*/Q, /*AG=*/P, /*HOUT=*/Q, as1, ad1, b1, g1, be1);

    // ---- classifier on first BATCH rows of h2 (Q) ----
    // z1 = h2[:B] @ Wc1 -> P ; relu(bn(z1 + bc1)) in place
    gemm_wmma_f32<HID, HID><<<dim3(HID / 128, BATCH / 32), 256, 0, stream>>>(Q, Wc1, P, BATCH);
    fill_f32<<<1, HID, 0, stream>>>(csum,   0.f, HID);
    fill_f32<<<1, HID, 0, stream>>>(csumsq, 0.f, HID);
    col_stats<<<(BATCH + 63) / 64, HID, 0, stream>>>(P, bc1, csum, csumsq, BATCH, HID, 0, 64);
    bn_finalize<<<1, HID, 0, stream>>>(csum, csumsq, cmean, crstd, BATCH, HID);
    bn_apply<<<(BATCH * HID + 255) / 256, 256, 0, stream>>>(P, bc1, gc1, bec1, cmean, crstd,
                                                            P, BATCH, HID, 0, 1);
    // z2 = c1 @ Wc2 (256->128) -> R ; relu(bn(z2 + bc2)) in place
    gemm_wmma_f32<128, HID><<<dim3(1, BATCH / 32), 256, 0, stream>>>(P, Wc2, R, BATCH);
    fill_f32<<<1, 128, 0, stream>>>(csum,   0.f, 128);
    fill_f32<<<1, 128, 0, stream>>>(csumsq, 0.f, 128);
    col_stats<<<(BATCH + 63) / 64, 128, 0, stream>>>(R, bc2, csum, csumsq, BATCH, 128, 0, 64);
    bn_finalize<<<1, 128, 0, stream>>>(csum, csumsq, cmean, crstd, BATCH, 128);
    bn_apply<<<(BATCH * 128 + 255) / 256, 256, 0, stream>>>(R, bc2, gc2, bec2, cmean, crstd,
                                                            R, BATCH, 128, 0, 1);
    // out = c2 @ Wc3 + bc3  (4096 scalars)
    final_dot<<<(BATCH * 32 + 255) / 256, 256, 0, stream>>>(R, Wc3, bc3, (float*)d_out, BATCH);
}